// Embedding_17635135717714
// MI455X (gfx1250) — compile-verified
//
#include <hip/hip_runtime.h>
#include <stdint.h>

// Embedding gather: out[b,h,:] = W[:, idx[b,h]]  -- pure data movement.
//   1) transpose W [64,100000] -> table [100000,64] in d_ws (25.6MB, L2-resident)
//   2) TDM gather-mode tensor_load_to_lds pulls 2x8 embedding rows (2x2KB) into
//      LDS; one normal-mode tensor_store_from_lds writes the contiguous 4KB
//      output region. Double-buffered per wave, throttled with s_wait_tensorcnt.
// Roofline: ~878MB total HBM traffic / 23.3 TB/s ~= 38us; everything else is
// about keeping descriptor/instruction overhead negligible versus that.

#define N_EMB  100000
#define DIM    64
#define BATCH  16384
#define HIST   200
#define ROWS   (BATCH * HIST)   /* 3,276,800 output rows of 64 floats  */
#define NPAIR  (ROWS / 16)      /* 204,800 iterations of 16 rows (exact) */

typedef __attribute__((ext_vector_type(4))) unsigned int v4u;
typedef __attribute__((ext_vector_type(4))) int          v4i;
typedef __attribute__((ext_vector_type(8))) int          v8i;

// ---- TDM builtin wrappers ---------------------------------------------------
// cpol must be an integer-constant expression -> non-type template parameter.
// This toolchain (clang>=23) takes the 6-arg form with a trailing v8i group.
template <int CPOL>
__device__ __forceinline__ void tdm_load_to_lds(v4u g0, v8i g1, v4i g2, v4i g3) {
#if __clang_major__ >= 23
  v8i gpad = {0, 0, 0, 0, 0, 0, 0, 0};
  __builtin_amdgcn_tensor_load_to_lds(g0, g1, g2, g3, gpad, CPOL);
#else
  __builtin_amdgcn_tensor_load_to_lds(g0, g1, g2, g3, CPOL);
#endif
}

template <int CPOL>
__device__ __forceinline__ void tdm_store_from_lds(v4u g0, v8i g1, v4i g2, v4i g3) {
#if __clang_major__ >= 23
  v8i gpad = {0, 0, 0, 0, 0, 0, 0, 0};
  __builtin_amdgcn_tensor_store_from_lds(g0, g1, g2, g3, gpad, CPOL);
#else
  __builtin_amdgcn_tensor_store_from_lds(g0, g1, g2, g3, CPOL);
#endif
}

__device__ __forceinline__ int rfl(int x) {
  return __builtin_amdgcn_readfirstlane(x);
}

// Issue one gather-mode TDM load: 8 rows of 64 f32 from `table` into LDS.
__device__ __forceinline__ void tdm_gather8(unsigned lds_off, uint64_t tbase,
                                            v4i idx_lo, v4i idx_hi) {
  v4u g0;
  g0.x = 0xC0000001u;                      // count=1 | gather_mode | 32-bit idx
  g0.y = lds_off;                          // LDS dest (bytes)
  g0.z = (unsigned)tbase;                  // global_addr[31:0]
  g0.w = (unsigned)((tbase >> 32) & 0x1FFFFFFull) | 0x80000000u; // [56:32]|type=2
  v8i g1 = {
    (int)0x00020000u,                      // data_size = 4B
    (int)(64u << 16),                      // tensor_dim0 = 64 (row width)
    (int)0x86A00000u,                      // tensor_dim1.lo16 = 100000 & 0xFFFF
    (int)((64u << 16) | 1u),               // tile_dim0=64 | tensor_dim1.hi16=1
    (int)8u,                               // tile_dim1 = #valid indices = 8
    (int)64u,                              // tensor_dim0_stride = 64 elements
    0, 0
  };
  tdm_load_to_lds</*cpol RT*/ 0>(g0, g1, idx_lo, idx_hi);
}

// Issue one contiguous 1-D TDM store of `nelem` f32 from LDS to `obase`.
__device__ __forceinline__ void tdm_store1d(unsigned lds_off, uint64_t obase,
                                            unsigned nelem) {
  v4u s0;
  s0.x = 1u;                               // count=1, normal mode
  s0.y = lds_off;
  s0.z = (unsigned)obase;
  s0.w = (unsigned)((obase >> 32) & 0x1FFFFFFull) | 0x80000000u;
  v8i s1 = {
    (int)0x00020000u,                      // data_size = 4B
    (int)(nelem << 16),                    // tensor_dim0 = nelem
    (int)0x00010000u,                      // tensor_dim1 = 1
    (int)(nelem << 16),                    // tile_dim0 = nelem (1-D tile)
    0,                                     // tile_dim1 = 0 (unused)
    (int)nelem,                            // dim0 stride (unused for 1-D)
    0, 0
  };
  v4i z = {0, 0, 0, 0};
  // TH=NT on the store: output is write-once/never-read; keep table hot in L2.
  tdm_store_from_lds</*cpol TH_NT*/ 1>(s0, s1, z, z);
}

// ---- Kernel: TDM gather (main path; defined first so disasm shows it) ------
__global__ void Emb_tdm_gather(const int* __restrict__ idx,
                               const float* __restrict__ table,
                               float* __restrict__ out) {
  __shared__ __align__(16) float lbuf[8 * 2 * 1024]; // 8 waves x 2 bufs x 4KB = 64KB
  lbuf[threadIdx.x] = 0.0f;   // make the LDS allocation concrete
  __syncthreads();

  const unsigned wave  = threadIdx.x >> 5;                  // 0..7
  const unsigned gwave = blockIdx.x * (blockDim.x >> 5) + wave;
  const unsigned nwave = gridDim.x * (blockDim.x >> 5);

  // generic-pointer low 32 bits == LDS byte offset (aperture lives in [63:32])
  const unsigned lds_base = (unsigned)(uintptr_t)(&lbuf[0]);
  const uint64_t tbase    = (uint64_t)(uintptr_t)table;

  unsigned it = 0;
  for (unsigned p = gwave; p < NPAIR; p += nwave, ++it) {
    const unsigned lds_off = lds_base + (wave * 2u + (it & 1u)) * 4096u;

    // 16 wave-uniform row indices for this pair of 8-row chunks
    const int* ip = idx + (size_t)p * 16;                   // 64B aligned
    const int4 a = *(const int4*)(ip);
    const int4 b = *(const int4*)(ip + 4);
    const int4 c = *(const int4*)(ip + 8);
    const int4 d = *(const int4*)(ip + 12);
    v4i i0 = {rfl(a.x), rfl(a.y), rfl(a.z), rfl(a.w)};
    v4i i1 = {rfl(b.x), rfl(b.y), rfl(b.z), rfl(b.w)};
    v4i i2 = {rfl(c.x), rfl(c.y), rfl(c.z), rfl(c.w)};
    v4i i3 = {rfl(d.x), rfl(d.y), rfl(d.z), rfl(d.w)};

    // Buffer reuse: store of iteration it-2 (same slot) must be done.
    // In-order TDM completion => allow {L,L,S} of iteration it-1 outstanding.
    __builtin_amdgcn_s_wait_tensorcnt(3);

    tdm_gather8(lds_off,         tbase, i0, i1);  // rows 16p   .. 16p+7
    tdm_gather8(lds_off + 2048u, tbase, i2, i3);  // rows 16p+8 .. 16p+15

    // One contiguous 4KB store covers both chunks (TDM ops of a wave are
    // processed in order, so the store reads LDS after both loads wrote it).
    const uint64_t obase = (uint64_t)(uintptr_t)(out + (size_t)p * 1024);
    tdm_store1d(lds_off, obase, 1024u);
  }
  // s_endpgm performs an implicit wait-idle (drains TENSORcnt)
}

// ---- Kernel: transpose W[64][100000] -> table[100000][64] ------------------
__global__ void Emb_transpose(const float* __restrict__ W,
                              float* __restrict__ table) {
  __shared__ __align__(16) float tile[64][68];  // stride 68 keeps rows 16B-aligned
  const int t  = threadIdx.x;        // 0..255
  const int i0 = blockIdx.x * 64;    // embedding tile base
  const int ti = t & 63;             // embedding within tile
  const int dg = t >> 6;             // 0..3 -> 16 d-values each
  const int i  = i0 + ti;
  if (i < N_EMB) {
#pragma unroll
    for (int k = 0; k < 16; ++k) {
      const int d = dg * 16 + k;
      tile[ti][d] = W[(size_t)d * N_EMB + i];  // coalesced along i
    }
  }
  __syncthreads();
  const int row = t >> 2;            // 0..63
  const int sb  = (t & 3) * 16;      // 0,16,32,48 (floats)
  const int gi  = i0 + row;
  if (gi < N_EMB) {
    float4* dst = (float4*)(table + (size_t)gi * DIM + sb);
#pragma unroll
    for (int k = 0; k < 4; ++k) {
      dst[k] = *(const float4*)&tile[row][sb + 4 * k];
    }
  }
}

// ---- Fallback: plain coalesced gather straight from W (if ws too small) ----
__global__ void Emb_direct_gather(const int* __restrict__ idx,
                                  const float* __restrict__ W,
                                  float* __restrict__ out) {
  const long long total  = (long long)ROWS * 32;
  const long long stride = (long long)gridDim.x * blockDim.x;
  for (long long g = (long long)blockIdx.x * blockDim.x + threadIdx.x;
       g < total; g += stride) {
    const int row = (int)(g >> 5);
    const int d0  = ((int)g & 31) * 2;
    const int e   = idx[row];
    float2 v;
    v.x = W[(size_t)d0 * N_EMB + e];
    v.y = W[(size_t)(d0 + 1) * N_EMB + e];
    *(float2*)(out + (size_t)row * DIM + d0) = v;   // coalesced 256B/row writes
  }
}

extern "C" void kernel_launch(void* const* d_in, const int* in_sizes, int n_in,
                              void* d_out, int out_size, void* d_ws, size_t ws_size,
                              hipStream_t stream) {
  const int*   indices = (const int*)d_in[0];   // [BATCH, HIST] int32
  const float* W       = (const float*)d_in[1]; // [DIM, N_EMB] f32
  float*       out     = (float*)d_out;         // [BATCH, HIST, DIM] f32
  (void)in_sizes; (void)n_in; (void)out_size;

  const size_t table_bytes = (size_t)N_EMB * DIM * sizeof(float);
  if (ws_size >= table_bytes) {
    float* table = (float*)d_ws;
    Emb_transpose<<<(N_EMB + 63) / 64, 256, 0, stream>>>(W, table);
    // 1024 blocks x 8 waves = 8192 waves -> exactly 25 pair-iterations each
    Emb_tdm_gather<<<1024, 256, 0, stream>>>(indices, table, out);
  } else {
    Emb_direct_gather<<<2048, 256, 0, stream>>>(indices, W, out);
  }
}